// TemporalConsistencyLoss_910533066825
// MI455X (gfx1250) — compile-verified
//
#include <hip/hip_runtime.h>

// TemporalConsistencyLoss — two (1024,192,78) fp32 tensors -> scalar loss.
// Memory-bound strided reduction; WMMA f32 16x16x4 (ones-B trick) used as the
// wave32 cross-lane reduction primitive.

typedef __attribute__((ext_vector_type(2))) float v2f;
typedef __attribute__((ext_vector_type(8))) float v8f;

#define TCL_ROWS (1024 * 192)   // B * N_PRIORS
#define TCL_FEAT 78
#define TCL_NBLK 512
#define TCL_NTHR 256

// Full 32-lane sum via one V_WMMA_F32_16X16X4_F32.
// B = all-ones => D[m][n] = sum_k A[m][k], independent of B's lane layout.
// A layout (ISA 7.12.2, 32-bit A 16x4): lane l<16 -> A[l][0]=v0, A[l][1]=v1;
// lane l>=16 -> A[l-16][2]=v0, A[l-16][3]=v1. With v1=0: rowsum(m)=p[m]+p[m+16].
// D layout: lanes 0..15 hold rows 0..7 in c[0..7]; lanes 16..31 hold rows 8..15.
// In-lane sum of c[0..7] + shfl_xor(16) => total over all 32 lanes, in every lane.
__device__ __forceinline__ float wave_sum_wmma(float p) {
    v2f a; a.x = p;    a.y = 0.0f;
    v2f b; b.x = 1.0f; b.y = 1.0f;
    v8f c = {};
    c = __builtin_amdgcn_wmma_f32_16x16x4_f32(
            /*neg_a=*/false, a, /*neg_b=*/false, b,
            /*c_mod=*/(short)0, c, /*reuse_a=*/false, /*reuse_b=*/false);
    float s = c[0] + c[1] + c[2] + c[3] + c[4] + c[5] + c[6] + c[7];
    s += __shfl_xor(s, 16, 32);
    return s;
}

__device__ __forceinline__ float smooth_l1(float d) {
    float a = fabsf(d);
    return (a < 1.0f) ? 0.5f * d * d : a - 0.5f;
}

__global__ __launch_bounds__(TCL_NTHR)
void tcl_partial_kernel(const float* __restrict__ cur,
                        const float* __restrict__ prv,
                        float* __restrict__ part) {
    __shared__ float lds[TCL_NTHR / 32][3];

    float cls = 0.0f, reg = 0.0f, fgs = 0.0f;

    // Each row needs only floats [0..5] of its 78-float record; 312-byte row
    // stride is 8B-aligned, so use three b64 loads per tensor per row.
    for (int row = blockIdx.x * blockDim.x + threadIdx.x; row < TCL_ROWS;
         row += gridDim.x * blockDim.x) {
        const float2* cr = (const float2*)(cur + (size_t)row * TCL_FEAT);
        const float2* pr = (const float2*)(prv + (size_t)row * TCL_FEAT);
        float2 c01 = cr[0], c23 = cr[1], c45 = cr[2];
        float2 p01 = pr[0], p23 = pr[1], p45 = pr[2];

        // 2-way softmax, current
        float cm   = fmaxf(c01.x, c01.y);
        float ce0  = __expf(c01.x - cm);
        float ce1  = __expf(c01.y - cm);
        float cinv = 1.0f / (ce0 + ce1);
        float cs0  = ce0 * cinv, cs1 = ce1 * cinv;
        // 2-way softmax, previous
        float pm   = fmaxf(p01.x, p01.y);
        float pe0  = __expf(p01.x - pm);
        float pe1  = __expf(p01.y - pm);
        float pinv = 1.0f / (pe0 + pe1);
        float ps0  = pe0 * pinv, ps1 = pe1 * pinv;

        float d0 = cs0 - ps0, d1 = cs1 - ps1;
        cls += d0 * d0 + d1 * d1;

        float fg = (ps1 > 0.1f) ? 1.0f : 0.0f;
        fgs += fg;

        float s = smooth_l1(c23.x - p23.x) + smooth_l1(c23.y - p23.y) +
                  smooth_l1(c45.x - p45.x) + smooth_l1(c45.y - p45.y);
        reg += s * 0.25f * fg;
    }

    // Wave-level reduction via WMMA (EXEC is all-ones here: uniform flow).
    float wc = wave_sum_wmma(cls);
    float wr = wave_sum_wmma(reg);
    float wf = wave_sum_wmma(fgs);

    const int lane = threadIdx.x & 31;
    const int wave = threadIdx.x >> 5;
    if (lane == 0) {
        lds[wave][0] = wc;
        lds[wave][1] = wr;
        lds[wave][2] = wf;
    }
    __syncthreads();

    if (threadIdx.x == 0) {
        float tc = 0.0f, tr = 0.0f, tf = 0.0f;
        for (int w = 0; w < TCL_NTHR / 32; ++w) {
            tc += lds[w][0];
            tr += lds[w][1];
            tf += lds[w][2];
        }
        part[blockIdx.x * 3 + 0] = tc;
        part[blockIdx.x * 3 + 1] = tr;
        part[blockIdx.x * 3 + 2] = tf;
    }
}

__global__ __launch_bounds__(32)
void tcl_final_kernel(const float* __restrict__ part, float* __restrict__ out) {
    float cls = 0.0f, reg = 0.0f, fgs = 0.0f;
    for (int i = threadIdx.x; i < TCL_NBLK; i += 32) {
        cls += part[i * 3 + 0];
        reg += part[i * 3 + 1];
        fgs += part[i * 3 + 2];
    }
    float tc = wave_sum_wmma(cls);
    float tr = wave_sum_wmma(reg);
    float tf = wave_sum_wmma(fgs);

    if (threadIdx.x == 0) {
        float cls_loss = tc / (float)(TCL_ROWS * 2) * 5.0f;
        float r        = tr / (tf + 1e-5f) * 5.0f;
        float reg_loss = (tf > 0.0f) ? r : 0.0f;
        out[0] = cls_loss + reg_loss;
    }
}

extern "C" void kernel_launch(void* const* d_in, const int* in_sizes, int n_in,
                              void* d_out, int out_size, void* d_ws, size_t ws_size,
                              hipStream_t stream) {
    const float* cur = (const float*)d_in[0];   // current_preds (1024,192,78) f32
    const float* prv = (const float*)d_in[1];   // previous_preds (1024,192,78) f32
    float* part = (float*)d_ws;                 // 512 * 3 floats = 6 KB scratch
    float* out  = (float*)d_out;                // 1 float

    tcl_partial_kernel<<<TCL_NBLK, TCL_NTHR, 0, stream>>>(cur, prv, part);
    tcl_final_kernel<<<1, 32, 0, stream>>>(part, out);
}